// TimeGraphNet_64183991272029
// MI455X (gfx1250) — compile-verified
//
#include <hip/hip_runtime.h>
#include <math.h>

typedef float v4f __attribute__((ext_vector_type(4)));
typedef float v2f __attribute__((ext_vector_type(2)));
typedef float v8f __attribute__((ext_vector_type(8)));

#define NSEQ 65536   // 64 batches * 1024 nodes
#define TLEN 1200
#define SPB  8       // sequences per block (one per wave32)

// ---------------------------------------------------------------------------
// Kernel 1: fused conv1->pool->conv2->pool->conv3 chain. One wave32 per
// sequence; sequence staged in LDS via coalesced non-temporal b128 loads
// (input is streamed exactly once; 315 MB => HBM-bound, ~13.5us floor).
// ---------------------------------------------------------------------------
__global__ __launch_bounds__(256) void tg_conv_chain(
    const float* __restrict__ in,    // [NSEQ, 1200]
    const float* __restrict__ w1, const float* __restrict__ b1,   // K=10
    const float* __restrict__ w2, const float* __restrict__ b2,   // K=3
    const float* __restrict__ w3, const float* __restrict__ b3,   // K=4
    float* __restrict__ feat)        // [NSEQ]
{
    // Row layout: s_in[w][3] = x[-1] pad, s_in[w][4+i] = x[i], s_in[w][1204] = x[1200] pad.
    // Row stride 1208 floats keeps &s_in[w][4] 16B-aligned for ds_store_b128.
    __shared__ float s_in[SPB][1208];
    __shared__ float s_p1[SPB][32];
    __shared__ float s_h2[SPB][12];
    __shared__ float s_p2[SPB][4];

    const int wave = threadIdx.x >> 5;
    const int lane = threadIdx.x & 31;
    const size_t seq = (size_t)blockIdx.x * SPB + wave;

    // ---- stage: 300 float4 per sequence, lane-interleaved, TH=NT ----
    const v4f* row = (const v4f*)(in + seq * TLEN);
    #pragma unroll
    for (int i = 0; i < 10; ++i) {
        int idx = i * 32 + lane;
        if (idx < 300) {
            v4f v = __builtin_nontemporal_load(row + idx);
            *(v4f*)&s_in[wave][4 + 4 * idx] = v;
        }
    }
    if (lane == 0) { s_in[wave][3] = 0.f; s_in[wave][1204] = 0.f; }

    // uniform weights -> scalar loads
    float cw1[10];
    #pragma unroll
    for (int k = 0; k < 10; ++k) cw1[k] = w1[k];
    const float cb1 = b1[0];
    float cw2[3];
    #pragma unroll
    for (int k = 0; k < 3; ++k) cw2[k] = w2[k];
    const float cb2 = b2[0];
    float cw3[4];
    #pragma unroll
    for (int k = 0; k < 4; ++k) cw3[k] = w3[k];
    const float cb3 = b3[0];

    __syncthreads();

    // ---- conv1 (K10,s8,p1) + ReLU + maxpool5 : lane l -> pooled output l (30 total)
    // conv out j reads padded idx 8j-1..8j+8 -> s_in idx 8j+3..8j+12.
    float p1 = 0.f;
    if (lane < 30) {
        const float* s = &s_in[wave][40 * lane + 3];
        p1 = -INFINITY;
        #pragma unroll
        for (int j = 0; j < 5; ++j) {
            float acc = cb1;
            #pragma unroll
            for (int k = 0; k < 10; ++k) acc = fmaf(cw1[k], s[8 * j + k], acc);
            p1 = fmaxf(p1, fmaxf(acc, 0.f));   // ReLU then pool (2nd ReLU is no-op)
        }
    }
    s_p1[wave][lane] = (lane < 30) ? p1 : 0.f;
    __syncthreads();

    // ---- conv2 (K3,s3) + ReLU : 10 outputs
    if (lane < 10) {
        float acc = cb2;
        #pragma unroll
        for (int k = 0; k < 3; ++k) acc = fmaf(cw2[k], s_p1[wave][3 * lane + k], acc);
        s_h2[wave][lane] = fmaxf(acc, 0.f);
    }
    __syncthreads();

    // ---- maxpool3 pad1 : 4 outputs (window 3m-1..3m+1, -inf pad)
    if (lane < 4) {
        float m = -INFINITY;
        #pragma unroll
        for (int k = -1; k <= 1; ++k) {
            int idx = 3 * lane + k;
            if (idx >= 0 && idx < 10) m = fmaxf(m, s_h2[wave][idx]);
        }
        s_p2[wave][lane] = m;
    }
    __syncthreads();

    // ---- conv3 (K4) + ReLU -> scalar feature
    if (lane == 0) {
        float acc = cb3;
        #pragma unroll
        for (int k = 0; k < 4; ++k) acc = fmaf(cw3[k], s_p2[wave][k], acc);
        feat[seq] = fmaxf(acc, 0.f);
    }
}

// ---------------------------------------------------------------------------
// Kernel 2: logits = feat[64,1024] @ cls_w.T[1024,3] + b, softmax.
// fp32 WMMA 16x16x4: 4 M-tiles of 16 rows, N padded 3->16, K=1024 split over
// 8 waves (128 each). Partials reduced in LDS, fused bias+softmax.
// A layout (16x4): lanes 0-15 hold K0/K1 in reg0/reg1, lanes 16-31 hold K2/K3.
// B layout (4x16): reg0 = rows K0 (lanes 0-15) / K2 (lanes 16-31), reg1 = K1/K3.
// ---------------------------------------------------------------------------
__global__ __launch_bounds__(256) void tg_cls_softmax(
    const float* __restrict__ feat,   // [64][1024]
    const float* __restrict__ cw,     // [3][1024]
    const float* __restrict__ cb,     // [3]
    float* __restrict__ out)          // [64][3]
{
    __shared__ float s_part[8][64][3];

    const int tid  = threadIdx.x;
    const int wv   = tid >> 5;        // 0..7: K slice
    const int lane = tid & 31;
    const int half = lane >> 4;       // 0 or 1
    const int lid  = lane & 15;

    v8f a0 = {}, a1 = {}, a2 = {}, a3 = {};

    const int   kbase  = wv * 128;
    const int   koff   = half * 2;
    const float bmask  = (lid < 3) ? 1.f : 0.f;                 // pad N 3->16 with zeros
    const float* brow  = cw + (size_t)((lid < 3) ? lid : 0) * 1024;  // always in-bounds

    const float* ar0 = feat + (size_t)( 0 + lid) * 1024;
    const float* ar1 = feat + (size_t)(16 + lid) * 1024;
    const float* ar2 = feat + (size_t)(32 + lid) * 1024;
    const float* ar3 = feat + (size_t)(48 + lid) * 1024;

    for (int kk = 0; kk < 128; kk += 4) {
        const int kb = kbase + kk + koff;      // this lane-half's K pair

        v2f B;
        B.x = brow[kb]     * bmask;
        B.y = brow[kb + 1] * bmask;

        v2f A0, A1, A2, A3;
        A0.x = ar0[kb]; A0.y = ar0[kb + 1];
        A1.x = ar1[kb]; A1.y = ar1[kb + 1];
        A2.x = ar2[kb]; A2.y = ar2[kb + 1];
        A3.x = ar3[kb]; A3.y = ar3[kb + 1];

        a0 = __builtin_amdgcn_wmma_f32_16x16x4_f32(false, A0, false, B, (short)0, a0, false, false);
        a1 = __builtin_amdgcn_wmma_f32_16x16x4_f32(false, A1, false, B, (short)0, a1, false, false);
        a2 = __builtin_amdgcn_wmma_f32_16x16x4_f32(false, A2, false, B, (short)0, a2, false, false);
        a3 = __builtin_amdgcn_wmma_f32_16x16x4_f32(false, A3, false, B, (short)0, a3, false, false);
    }

    // C/D layout: VGPR r, lanes 0-15 -> M=r, lanes 16-31 -> M=r+8; N = lid.
    if (lid < 3) {
        #pragma unroll
        for (int r = 0; r < 8; ++r) {
            const int m = 8 * half + r;
            s_part[wv][ 0 + m][lid] = a0[r];
            s_part[wv][16 + m][lid] = a1[r];
            s_part[wv][32 + m][lid] = a2[r];
            s_part[wv][48 + m][lid] = a3[r];
        }
    }
    __syncthreads();

    if (tid < 64) {
        float l0 = cb[0], l1 = cb[1], l2 = cb[2];
        #pragma unroll
        for (int w = 0; w < 8; ++w) {
            l0 += s_part[w][tid][0];
            l1 += s_part[w][tid][1];
            l2 += s_part[w][tid][2];
        }
        const float mx = fmaxf(l0, fmaxf(l1, l2));
        const float e0 = expf(l0 - mx), e1 = expf(l1 - mx), e2 = expf(l2 - mx);
        const float inv = 1.f / (e0 + e1 + e2);
        out[tid * 3 + 0] = e0 * inv;
        out[tid * 3 + 1] = e1 * inv;
        out[tid * 3 + 2] = e2 * inv;
    }
}

extern "C" void kernel_launch(void* const* d_in, const int* in_sizes, int n_in,
                              void* d_out, int out_size, void* d_ws, size_t ws_size,
                              hipStream_t stream) {
    const float* in = (const float*)d_in[0];
    const float* w1 = (const float*)d_in[1];
    const float* b1 = (const float*)d_in[2];
    const float* w2 = (const float*)d_in[3];
    const float* b2 = (const float*)d_in[4];
    const float* w3 = (const float*)d_in[5];
    const float* b3 = (const float*)d_in[6];
    // d_in[7..10] = gcn weights: dead code in the reference
    const float* cw = (const float*)d_in[11];
    const float* cb = (const float*)d_in[12];

    float* feat = (float*)d_ws;   // 65536 floats = 256 KB scratch

    tg_conv_chain<<<NSEQ / SPB, 256, 0, stream>>>(in, w1, b1, w2, b2, w3, b3, feat);
    tg_cls_softmax<<<1, 256, 0, stream>>>(feat, cw, cb, (float*)d_out);
}